// DeeperGCN_1726576853643
// MI455X (gfx1250) — compile-verified
//
#include <hip/hip_runtime.h>
#include <hip/hip_bf16.h>
#include <math.h>

// DeeperGCN forward for MI455X (gfx1250, wave32).
// - fp32 WMMA (v_wmma_f32_16x16x4_f32) for encoders + per-layer MLPs
// - CSR-by-dst segment softmax (no float atomics, deterministic)
// - eh stored bf16 to halve the dominant HBM stream (8x E*64 reads)

typedef float v2f __attribute__((ext_vector_type(2)));
typedef float v8f __attribute__((ext_vector_type(8)));

#define HDIM 64
#define HHDIM 128
#define LN_EPS 1e-5f
#define MSG_EPS_C 1e-7f
#define SM_EPS_C 1e-16f

static __device__ __forceinline__ float bf2f(unsigned short h) {
  return __uint_as_float(((unsigned)h) << 16);
}
static __device__ __forceinline__ unsigned short f2bf(float f) {
  unsigned u = __float_as_uint(f);
  u += 0x7FFFu + ((u >> 16) & 1u);  // round-to-nearest-even
  return (unsigned short)(u >> 16);
}
static __device__ __forceinline__ float wredsum(float v) {
#pragma unroll
  for (int m = 16; m >= 1; m >>= 1) v += __shfl_xor(v, m, 32);
  return v;
}

// ---------------- CSR build ----------------
__global__ void k_zero_i32(int* p, int n) {
  int i = blockIdx.x * blockDim.x + threadIdx.x;
  if (i < n) p[i] = 0;
}
__global__ void k_deg(const int* __restrict__ dstv, int* __restrict__ deg, int E) {
  int e = blockIdx.x * blockDim.x + threadIdx.x;
  if (e < E) atomicAdd(&deg[dstv[e]], 1);
}
__global__ void k_scan(const int* __restrict__ deg, int* __restrict__ rowptr, int n) {
  __shared__ int buf[1024];
  __shared__ int carry;
  int tid = threadIdx.x;
  if (tid == 0) { carry = 0; rowptr[0] = 0; }
  __syncthreads();
  for (int base = 0; base < n; base += 1024) {
    int i = base + tid;
    buf[tid] = (i < n) ? deg[i] : 0;
    __syncthreads();
    for (int off = 1; off < 1024; off <<= 1) {
      int add = (tid >= off) ? buf[tid - off] : 0;
      __syncthreads();
      buf[tid] += add;
      __syncthreads();
    }
    if (i < n) rowptr[i + 1] = carry + buf[tid];
    __syncthreads();
    if (tid == 0) carry += buf[1023];
    __syncthreads();
  }
}
__global__ void k_copy_i32(const int* __restrict__ a, int* __restrict__ b, int n) {
  int i = blockIdx.x * blockDim.x + threadIdx.x;
  if (i < n) b[i] = a[i];
}
__global__ void k_csr_fill(const int* __restrict__ srcv, const int* __restrict__ dstv,
                           int* __restrict__ cursor, int* __restrict__ csrc,
                           int* __restrict__ ceid, int E) {
  int e = blockIdx.x * blockDim.x + threadIdx.x;
  if (e < E) {
    int d = dstv[e];
    int p = atomicAdd(&cursor[d], 1);
    csrc[p] = srcv[e];
    ceid[p] = e;
  }
}
// per-node insertion sort by edge id -> bitwise-deterministic reductions
__global__ void k_csr_sort(int* __restrict__ eid, int* __restrict__ srcv,
                           const int* __restrict__ rowptr, int n) {
  int i = blockIdx.x * blockDim.x + threadIdx.x;
  if (i >= n) return;
  int r0 = rowptr[i], r1 = rowptr[i + 1];
  for (int j = r0 + 1; j < r1; ++j) {
    int ke = eid[j], ks = srcv[j];
    int k = j - 1;
    while (k >= r0 && eid[k] > ke) {
      eid[k + 1] = eid[k];
      srcv[k + 1] = srcv[k];
      --k;
    }
    eid[k + 1] = ke;
    srcv[k + 1] = ks;
  }
}

// ---------------- WMMA encoders ----------------
// eh[E,64] = bf16(edge_attr[E,16] @ ee_w[16,64] + ee_b). grid = E/16, block = 128 (4 waves).
__global__ void k_edge_encode(const float* __restrict__ ea, const float* __restrict__ w,
                              const float* __restrict__ bias, unsigned short* __restrict__ ehbf) {
  int wave = threadIdx.x >> 5, lane = threadIdx.x & 31;
  int hf = lane >> 4, idx = lane & 15, kb = hf * 2;
  int rowBase = blockIdx.x * 16;
  int m = rowBase + idx;
  int col = wave * 16 + idx;
  v8f acc = {};
#pragma unroll
  for (int ks = 0; ks < 4; ++ks) {
    int k = ks * 4 + kb;
    v2f a, b;
    a.x = ea[m * 16 + k];     a.y = ea[m * 16 + k + 1];
    b.x = w[k * HDIM + col];  b.y = w[(k + 1) * HDIM + col];
    acc = __builtin_amdgcn_wmma_f32_16x16x4_f32(false, a, false, b, (short)0, acc, false, false);
  }
  float bs = bias[col];
#pragma unroll
  for (int r = 0; r < 8; ++r) {
    int row = rowBase + hf * 8 + r;
    ehbf[(size_t)row * HDIM + col] = f2bf(acc[r] + bs);
  }
}

// xh[N,64] = x[N,32] @ enc_w[32,64] + enc_b. grid = N/16, block = 128.
__global__ void k_node_encode(const float* __restrict__ x, const float* __restrict__ w,
                              const float* __restrict__ bias, float* __restrict__ xh) {
  int wave = threadIdx.x >> 5, lane = threadIdx.x & 31;
  int hf = lane >> 4, idx = lane & 15, kb = hf * 2;
  int rowBase = blockIdx.x * 16;
  int m = rowBase + idx;
  int col = wave * 16 + idx;
  v8f acc = {};
#pragma unroll
  for (int ks = 0; ks < 8; ++ks) {
    int k = ks * 4 + kb;
    v2f a, b;
    a.x = x[m * 32 + k];      a.y = x[m * 32 + k + 1];
    b.x = w[k * HDIM + col];  b.y = w[(k + 1) * HDIM + col];
    acc = __builtin_amdgcn_wmma_f32_16x16x4_f32(false, a, false, b, (short)0, acc, false, false);
  }
  float bs = bias[col];
#pragma unroll
  for (int r = 0; r < 8; ++r) {
    int row = rowBase + hf * 8 + r;
    xh[(size_t)row * HDIM + col] = acc[r] + bs;
  }
}

// ---------------- elementwise: relu(LayerNorm(x)) over H=64, wave per row ----------------
__global__ void k_norm_relu(const float* __restrict__ x, const float* __restrict__ g,
                            const float* __restrict__ b, float* __restrict__ out, int n) {
  int row = blockIdx.x * 8 + (threadIdx.x >> 5);
  int lane = threadIdx.x & 31;
  if (row >= n) return;
  float2 v = *reinterpret_cast<const float2*>(x + (size_t)row * HDIM + lane * 2);
  float s1 = wredsum(v.x + v.y);
  float s2 = wredsum(v.x * v.x + v.y * v.y);
  float mean = s1 * (1.0f / HDIM);
  float inv = rsqrtf(s2 * (1.0f / HDIM) - mean * mean + LN_EPS);
  float2 o;
  o.x = fmaxf((v.x - mean) * inv * g[lane * 2] + b[lane * 2], 0.0f);
  o.y = fmaxf((v.y - mean) * inv * g[lane * 2 + 1] + b[lane * 2 + 1], 0.0f);
  *reinterpret_cast<float2*>(out + (size_t)row * HDIM + lane * 2) = o;
}

// ---------------- GENConv segment softmax: wave per dst node, lane = 2 features ----------------
__global__ void k_conv(const float* __restrict__ hin, const unsigned short* __restrict__ ehbf,
                       const int* __restrict__ rowptr, const int* __restrict__ csrc,
                       const int* __restrict__ ceid, const float* __restrict__ tp,
                       float* __restrict__ hsum, int n) {
  int node = blockIdx.x * 8 + (threadIdx.x >> 5);
  int lane = threadIdx.x & 31;
  if (node >= n) return;
  int f = lane * 2;
  int r0 = rowptr[node], r1 = rowptr[node + 1];
  float t = tp[0];
  float mx0 = -INFINITY, mx1 = -INFINITY;
  for (int j = r0; j < r1; ++j) {
    int s = csrc[j], e = ceid[j];
    float2 hv = *reinterpret_cast<const float2*>(hin + (size_t)s * HDIM + f);
    unsigned ev = *reinterpret_cast<const unsigned*>(ehbf + (size_t)e * HDIM + f);
    float m0 = fmaxf(hv.x + bf2f((unsigned short)(ev & 0xffffu)), 0.0f) + MSG_EPS_C;
    float m1 = fmaxf(hv.y + bf2f((unsigned short)(ev >> 16)), 0.0f) + MSG_EPS_C;
    mx0 = fmaxf(mx0, m0 * t);
    mx1 = fmaxf(mx1, m1 * t);
  }
  float den0 = 0.f, den1 = 0.f, num0 = 0.f, num1 = 0.f;
  for (int j = r0; j < r1; ++j) {
    int s = csrc[j], e = ceid[j];
    float2 hv = *reinterpret_cast<const float2*>(hin + (size_t)s * HDIM + f);
    unsigned ev = *reinterpret_cast<const unsigned*>(ehbf + (size_t)e * HDIM + f);
    float m0 = fmaxf(hv.x + bf2f((unsigned short)(ev & 0xffffu)), 0.0f) + MSG_EPS_C;
    float m1 = fmaxf(hv.y + bf2f((unsigned short)(ev >> 16)), 0.0f) + MSG_EPS_C;
    float e0 = expf(m0 * t - mx0);
    float e1 = expf(m1 * t - mx1);
    den0 += e0; den1 += e1;
    num0 += m0 * e0; num1 += m1 * e1;
  }
  float2 xv = *reinterpret_cast<const float2*>(hin + (size_t)node * HDIM + f);
  float2 o;
  o.x = num0 / (den0 + SM_EPS_C) + xv.x;
  o.y = num1 / (den1 + SM_EPS_C) + xv.y;
  *reinterpret_cast<float2*>(hsum + (size_t)node * HDIM + f) = o;
}

// ---------------- fused MLP: Lin(64->128) + LN + ReLU + Lin(128->64), 16 rows/block ----------------
__global__ void k_mlp(const float* __restrict__ hs, const float* __restrict__ w1,
                      const float* __restrict__ b1, const float* __restrict__ lg,
                      const float* __restrict__ lb, const float* __restrict__ w2,
                      const float* __restrict__ b2, float* __restrict__ xh, int beta) {
  __shared__ float u[16 * 132];  // 132 stride kills LDS bank conflicts on column reads
  int wave = threadIdx.x >> 5, lane = threadIdx.x & 31;
  int hf = lane >> 4, idx = lane & 15, kb = hf * 2;
  int rowBase = blockIdx.x * 16;
  // GEMM1: [16,64] @ [64,128]; 8 col-tiles, 2 per wave
#pragma unroll
  for (int cc = 0; cc < 2; ++cc) {
    int col = (wave + cc * 4) * 16 + idx;
    v8f acc = {};
#pragma unroll
    for (int ks = 0; ks < 16; ++ks) {
      int k = ks * 4 + kb;
      v2f a, b;
      a.x = hs[(size_t)(rowBase + idx) * HDIM + k];
      a.y = hs[(size_t)(rowBase + idx) * HDIM + k + 1];
      b.x = w1[k * HHDIM + col];
      b.y = w1[(k + 1) * HHDIM + col];
      acc = __builtin_amdgcn_wmma_f32_16x16x4_f32(false, a, false, b, (short)0, acc, false, false);
    }
    float bs = b1[col];
#pragma unroll
    for (int r = 0; r < 8; ++r) u[(hf * 8 + r) * 132 + col] = acc[r] + bs;
  }
  __syncthreads();
  // LayerNorm(128) + ReLU, wave handles 4 rows, lane handles 4 cols
#pragma unroll
  for (int rr = 0; rr < 4; ++rr) {
    int row = wave * 4 + rr;
    float s1 = 0.f, s2 = 0.f;
#pragma unroll
    for (int q = 0; q < 4; ++q) {
      float x = u[row * 132 + lane + q * 32];
      s1 += x; s2 += x * x;
    }
    s1 = wredsum(s1); s2 = wredsum(s2);
    float mean = s1 * (1.0f / HHDIM);
    float inv = rsqrtf(s2 * (1.0f / HHDIM) - mean * mean + LN_EPS);
#pragma unroll
    for (int q = 0; q < 4; ++q) {
      int c = lane + q * 32;
      float x = (u[row * 132 + c] - mean) * inv * lg[c] + lb[c];
      u[row * 132 + c] = fmaxf(x, 0.0f);
    }
  }
  __syncthreads();
  // GEMM2: [16,128] @ [128,64]; 4 col-tiles, 1 per wave
  {
    int col = wave * 16 + idx;
    v8f acc = {};
#pragma unroll
    for (int ks = 0; ks < 32; ++ks) {
      int k = ks * 4 + kb;
      v2f a, b;
      a.x = u[idx * 132 + k];
      a.y = u[idx * 132 + k + 1];
      b.x = w2[k * HDIM + col];
      b.y = w2[(k + 1) * HDIM + col];
      acc = __builtin_amdgcn_wmma_f32_16x16x4_f32(false, a, false, b, (short)0, acc, false, false);
    }
    float bs = b2[col];
#pragma unroll
    for (int r = 0; r < 8; ++r) {
      size_t o = (size_t)(rowBase + hf * 8 + r) * HDIM + col;
      float val = acc[r] + bs;
      if (beta) xh[o] += val; else xh[o] = val;
    }
  }
}

// ---------------- final LN + ReLU + Linear(64->1), wave per row ----------------
__global__ void k_final(const float* __restrict__ xh, const float* __restrict__ g,
                        const float* __restrict__ b, const float* __restrict__ lw,
                        const float* __restrict__ lbias, float* __restrict__ outn, int n) {
  int row = blockIdx.x * 8 + (threadIdx.x >> 5);
  int lane = threadIdx.x & 31;
  if (row >= n) return;
  float2 v = *reinterpret_cast<const float2*>(xh + (size_t)row * HDIM + lane * 2);
  float s1 = wredsum(v.x + v.y);
  float s2 = wredsum(v.x * v.x + v.y * v.y);
  float mean = s1 * (1.0f / HDIM);
  float inv = rsqrtf(s2 * (1.0f / HDIM) - mean * mean + LN_EPS);
  float h0 = fmaxf((v.x - mean) * inv * g[lane * 2] + b[lane * 2], 0.0f);
  float h1 = fmaxf((v.y - mean) * inv * g[lane * 2 + 1] + b[lane * 2 + 1], 0.0f);
  float d = wredsum(h0 * lw[lane * 2] + h1 * lw[lane * 2 + 1]);
  if (lane == 0) outn[row] = d + lbias[0];
}

// ---------------- deterministic mean-pool: batch is sorted, binary-search ranges ----------------
__global__ void k_pool(const float* __restrict__ outn, const int* __restrict__ batch,
                       float* __restrict__ out, int n) {
  int gph = blockIdx.x;
  int lo, hi;
  { int a = 0, bb = n; while (a < bb) { int mid = (a + bb) >> 1; if (batch[mid] < gph) a = mid + 1; else bb = mid; } lo = a; }
  { int a = lo, bb = n; while (a < bb) { int mid = (a + bb) >> 1; if (batch[mid] < gph + 1) a = mid + 1; else bb = mid; } hi = a; }
  __shared__ float red[256];
  float s = 0.f;
  for (int i = lo + (int)threadIdx.x; i < hi; i += 256) s += outn[i];
  red[threadIdx.x] = s;
  __syncthreads();
  for (int off = 128; off >= 1; off >>= 1) {
    if ((int)threadIdx.x < off) red[threadIdx.x] += red[threadIdx.x + off];
    __syncthreads();
  }
  if (threadIdx.x == 0) {
    int cnt = hi - lo;
    out[gph] = red[0] / (float)(cnt > 0 ? cnt : 1);
  }
}

extern "C" void kernel_launch(void* const* d_in, const int* in_sizes, int n_in,
                              void* d_out, int out_size, void* d_ws, size_t ws_size,
                              hipStream_t stream) {
  const float* x     = (const float*)d_in[0];
  const int*   ei    = (const int*)d_in[1];
  const float* ea    = (const float*)d_in[2];
  const int*   batch = (const int*)d_in[3];
  const float* enc_w = (const float*)d_in[4];
  const float* enc_b = (const float*)d_in[5];
  const float* ee_w  = (const float*)d_in[6];
  const float* ee_b  = (const float*)d_in[7];
  const float* w1    = (const float*)d_in[8];
  const float* b1    = (const float*)d_in[9];
  const float* lng   = (const float*)d_in[10];
  const float* lnb   = (const float*)d_in[11];
  const float* w2    = (const float*)d_in[12];
  const float* b2    = (const float*)d_in[13];
  const float* tv    = (const float*)d_in[14];
  const float* ng    = (const float*)d_in[15];
  const float* nb    = (const float*)d_in[16];
  const float* lin_w = (const float*)d_in[17];
  const float* lin_b = (const float*)d_in[18];
  float* out = (float*)d_out;

  const int N = in_sizes[0] / 32;   // 50000 (multiple of 16)
  const int E = in_sizes[2] / 16;   // 800000 (multiple of 16)
  const int G = out_size;           // 256
  const int L = in_sizes[14];       // 4
  const int* srcv = ei;
  const int* dstv = ei + E;

  char* p = (char*)d_ws;
  auto carve = [&](size_t bytes) -> void* {
    void* r = (void*)p;
    p += (bytes + 255) & ~(size_t)255;
    return r;
  };
  unsigned short* ehbf = (unsigned short*)carve((size_t)E * HDIM * 2);
  float* xh    = (float*)carve((size_t)N * HDIM * 4);
  float* hbuf  = (float*)carve((size_t)N * HDIM * 4);
  float* hsum  = (float*)carve((size_t)N * HDIM * 4);
  float* outn  = (float*)carve((size_t)N * 4);
  int* rowptr  = (int*)carve((size_t)(N + 1) * 4);
  int* deg     = (int*)carve((size_t)N * 4);
  int* cursor  = (int*)carve((size_t)N * 4);
  int* csrc    = (int*)carve((size_t)E * 4);
  int* ceid    = (int*)carve((size_t)E * 4);
  (void)ws_size; (void)n_in;

  // CSR by dst (rebuilt every call; deterministic after sort)
  k_zero_i32<<<(N + 255) / 256, 256, 0, stream>>>(deg, N);
  k_deg<<<(E + 255) / 256, 256, 0, stream>>>(dstv, deg, E);
  k_scan<<<1, 1024, 0, stream>>>(deg, rowptr, N);
  k_copy_i32<<<(N + 255) / 256, 256, 0, stream>>>(rowptr, cursor, N);
  k_csr_fill<<<(E + 255) / 256, 256, 0, stream>>>(srcv, dstv, cursor, csrc, ceid, E);
  k_csr_sort<<<(N + 255) / 256, 256, 0, stream>>>(ceid, csrc, rowptr, N);

  // Encoders (WMMA)
  k_edge_encode<<<E / 16, 128, 0, stream>>>(ea, ee_w, ee_b, ehbf);
  k_node_encode<<<N / 16, 128, 0, stream>>>(x, enc_w, enc_b, xh);

  for (int i = 0; i < L; ++i) {
    const float* hin = xh;
    if (i > 0) {
      k_norm_relu<<<(N + 7) / 8, 256, 0, stream>>>(xh, ng + i * HDIM, nb + i * HDIM, hbuf, N);
      hin = hbuf;
    }
    k_conv<<<(N + 7) / 8, 256, 0, stream>>>(hin, ehbf, rowptr, csrc, ceid, tv + i, hsum, N);
    k_mlp<<<N / 16, 128, 0, stream>>>(hsum, w1 + (size_t)i * HDIM * HHDIM, b1 + i * HHDIM,
                                      lng + i * HHDIM, lnb + i * HHDIM,
                                      w2 + (size_t)i * HHDIM * HDIM, b2 + i * HDIM,
                                      xh, i > 0 ? 1 : 0);
  }

  k_final<<<(N + 7) / 8, 256, 0, stream>>>(xh, ng, nb, lin_w, lin_b, outn, N);
  k_pool<<<G, 256, 0, stream>>>(outn, batch, out, N);
}